// GRUEncoder_42674795053087
// MI455X (gfx1250) — compile-verified
//
#include <hip/hip_runtime.h>

#define T_LEN 524288
#define HID 20
#define G3 60
#define XG_STRIDE 64
#define UNROLL 8

typedef __attribute__((ext_vector_type(2))) float v2f;
typedef __attribute__((ext_vector_type(8))) float v8f;

// ---------------------------------------------------------------------------
// Phase 1: batched input projection  XG[t][n] = x[t]*W_ih[n] + b_ih[n]
// (mirrors the reference's precompute "xt = x @ W_ih.T + b_ih")
// Done with V_WMMA_F32_16X16X4_F32:
//   A[m,k] = x[t0+m] at k==0 only  (A layout documented: lanes0-15 vgpr0=K0)
//   B[k,n] = W_ih[n] for ALL k     (robust to B's exact K-slot mapping,
//                                   since A is nonzero only at K=0)
//   C[m,n] = b_ih[n]
// Each wave produces a 16-timestep x 64-column tile with 4 WMMAs.
// ---------------------------------------------------------------------------
__global__ __launch_bounds__(256) void xg_wmma_kernel(const float* __restrict__ x,
                                                      const float* __restrict__ Wih,
                                                      const float* __restrict__ bih,
                                                      float* __restrict__ XG) {
    const int lane = threadIdx.x & 31;
    const int wave = threadIdx.x >> 5;
    const int t0   = (blockIdx.x * 8 + wave) * 16;
    const int hl   = lane & 15;

    v2f a;
    a.x = (lane < 16) ? x[t0 + hl] : 0.0f;  // lanes0-15 vgpr0 = K0 ; lanes16-31 = K2 (zero)
    a.y = 0.0f;                             // K1 / K3 = zero

    #pragma unroll
    for (int nt = 0; nt < 4; ++nt) {
        const int n   = nt * 16 + hl;
        const float w = (n < G3) ? Wih[n] : 0.0f;
        const float bb = (n < G3) ? bih[n] : 0.0f;
        v2f b; b.x = w; b.y = w;            // every K slot = W_ih[n]
        v8f c;
        #pragma unroll
        for (int i = 0; i < 8; ++i) c[i] = bb;

        v8f d = __builtin_amdgcn_wmma_f32_16x16x4_f32(
            /*neg_a=*/false, a, /*neg_b=*/false, b,
            /*c_mod=*/(short)0, c, /*reuse_a=*/false, /*reuse_b=*/false);

        #pragma unroll
        for (int i = 0; i < 8; ++i) {
            const int m = (lane < 16) ? i : (8 + i);  // D layout: vgpr i -> M=i / M=8+i
            XG[(size_t)(t0 + m) * XG_STRIDE + n] = d[i];
        }
    }
}

// ---------------------------------------------------------------------------
// Phase 2: the serial scan. One wave32; lane j owns hidden unit j.
// h is broadcast through SGPRs via v_readlane, so the 60 recurrent FMAs per
// step use scalar-source v_fmac. XG loads are software-pipelined UNROLL steps
// ahead (XG is L2-resident: 134 MB < 192 MB L2).
// ---------------------------------------------------------------------------
__device__ __forceinline__ float fast_sig(float v) {
    // sigmoid(v) = 1 / (1 + 2^(-v*log2e))
    float e = __builtin_amdgcn_exp2f(v * -1.4426950408889634f);
    return __builtin_amdgcn_rcpf(1.0f + e);
}

template <bool USE_XG>
__global__ __launch_bounds__(32) void gru_scan_kernel(const float* __restrict__ x,
                                                      const float* __restrict__ Wih,
                                                      const float* __restrict__ Whh,
                                                      const float* __restrict__ bih,
                                                      const float* __restrict__ bhh,
                                                      const float* __restrict__ XG,
                                                      float* __restrict__ out) {
    const int lane = threadIdx.x & 31;
    const int j = (lane < HID) ? lane : 0;   // lanes 20..31 shadow lane 0 (harmless)

    // Per-lane recurrent weight rows (r,z,n gate rows of W_hh; torch order).
    float wr[HID], wz[HID], wn[HID];
    #pragma unroll
    for (int k = 0; k < HID; ++k) {
        wr[k] = Whh[(0 * HID + j) * HID + k];
        wz[k] = Whh[(1 * HID + j) * HID + k];
        wn[k] = Whh[(2 * HID + j) * HID + k];
    }
    const float bhr = bhh[j], bhz = bhh[HID + j], bhn = bhh[2 * HID + j];

    float wir = 0.f, wiz = 0.f, win = 0.f, bir = 0.f, biz = 0.f, binn = 0.f;
    if (!USE_XG) {
        wir = Wih[j];  wiz = Wih[HID + j];  win = Wih[2 * HID + j];
        bir = bih[j];  biz = bih[HID + j];  binn = bih[2 * HID + j];
    }

    // h broadcast as wave-uniform values (SGPRs after readlane).
    float sh[HID];
    #pragma unroll
    for (int k = 0; k < HID; ++k) sh[k] = 0.0f;

    // Software-pipelined prefetch buffers.
    float pr[UNROLL], pz[UNROLL], pn[UNROLL];
    const float* xgp = USE_XG ? (XG + j) : nullptr;
    if (USE_XG) {
        #pragma unroll
        for (int u = 0; u < UNROLL; ++u) {
            const size_t base = (size_t)u * XG_STRIDE;
            pr[u] = xgp[base];
            pz[u] = xgp[base + HID];
            pn[u] = xgp[base + 2 * HID];
        }
    } else {
        #pragma unroll
        for (int u = 0; u < UNROLL; ++u) pr[u] = x[u];
    }

    float hlane = 0.0f;  // h[j] lives in lane j

    for (int t = 0; t < T_LEN; t += UNROLL) {
        #pragma unroll
        for (int u = 0; u < UNROLL; ++u) {
            float xr, xz, xn;
            if (USE_XG) {
                xr = pr[u]; xz = pz[u]; xn = pn[u];
                const size_t nb = (size_t)(t + u + UNROLL) * XG_STRIDE;  // ws over-allocated
                pr[u] = xgp[nb];
                pz[u] = xgp[nb + HID];
                pn[u] = xgp[nb + 2 * HID];
            } else {
                const float xt = pr[u];
                int nf = t + u + UNROLL;
                if (nf > T_LEN - 1) nf = T_LEN - 1;
                pr[u] = x[nf];
                xr = fmaf(xt, wir, bir);
                xz = fmaf(xt, wiz, biz);
                xn = fmaf(xt, win, binn);
            }

            // hg = W_hh @ h + b_hh   (60 scalar-source FMAs, 3-way ILP)
            float ar = bhr, az = bhz, an = bhn;
            #pragma unroll
            for (int k = 0; k < HID; ++k) {
                ar = fmaf(sh[k], wr[k], ar);
                az = fmaf(sh[k], wz[k], az);
                an = fmaf(sh[k], wn[k], an);
            }
            const float r = fast_sig(xr + ar);
            const float z = fast_sig(xz + az);
            const float nv = fmaf(r, an, xn);
            const float n = fmaf(2.0f, fast_sig(nv + nv), -1.0f);  // tanh(nv)
            hlane = fmaf(z, hlane - n, n);                          // (1-z)*n + z*h

            // Broadcast new h to wave-uniform scalars for the next step.
            #pragma unroll
            for (int k = 0; k < HID; ++k)
                sh[k] = __int_as_float(__builtin_amdgcn_readlane(__float_as_int(hlane), k));
        }
    }

    if (lane < HID) out[lane] = hlane;
}

// ---------------------------------------------------------------------------
extern "C" void kernel_launch(void* const* d_in, const int* in_sizes, int n_in,
                              void* d_out, int out_size, void* d_ws, size_t ws_size,
                              hipStream_t stream) {
    const float* x   = (const float*)d_in[0];
    const float* Wih = (const float*)d_in[1];
    const float* Whh = (const float*)d_in[2];
    const float* bih = (const float*)d_in[3];
    const float* bhh = (const float*)d_in[4];
    float* out = (float*)d_out;

    const size_t need = (size_t)(T_LEN + UNROLL) * XG_STRIDE * sizeof(float);  // ~128 MiB
    if (ws_size >= need) {
        float* XG = (float*)d_ws;
        // 256 threads = 8 waves; each wave: 16 timesteps x 64 cols via 4 WMMAs.
        xg_wmma_kernel<<<T_LEN / 128, 256, 0, stream>>>(x, Wih, bih, XG);
        gru_scan_kernel<true><<<1, 32, 0, stream>>>(x, Wih, Whh, bih, bhh, XG, out);
    } else {
        gru_scan_kernel<false><<<1, 32, 0, stream>>>(x, Wih, Whh, bih, bhh, nullptr, out);
    }
}